// EMDLoss_75333726372146
// MI455X (gfx1250) — compile-verified
//
#include <hip/hip_runtime.h>
#include <math.h>

// ---------------- problem constants (from reference) ----------------
#define BATCH 4
#define NPTS  2048
#define TILE  16
#define NT    128                      // NPTS / TILE
#define EPS_C 0.1f
#define NITS  50
#define TOL_C 1e-3f
#define WLOG  (-1.3862943611198906f)   // log(1/4)
#define SRT   4.47213595499957939f     // sqrt(2/eps) = sqrt(20)

typedef __attribute__((ext_vector_type(2))) float v2f;
typedef __attribute__((ext_vector_type(8))) float v8f;

static __device__ __forceinline__ v8f wmma4(v2f a, v2f b) {
    v8f c = {};
    return __builtin_amdgcn_wmma_f32_16x16x4_f32(false, a, false, b, (short)0, c,
                                                 false, false);
}
static __device__ __forceinline__ float max8(v8f d) {
    return fmaxf(fmaxf(fmaxf(d[0], d[1]), fmaxf(d[2], d[3])),
                 fmaxf(fmaxf(d[4], d[5]), fmaxf(d[6], d[7])));
}

// ---------------------------------------------------------------
// prep: WMMA-native interleaved packing per point:
//   pack[2i]   = (s*x0, s*x1)            (lanes 0-15: K0,K1)
//   pack[2i+1] = (s*x2, wlog - n)        (lanes 16-31: K2,K3),  n = ||x||^2/eps
// ---------------------------------------------------------------
__global__ void prep_kernel(const float* __restrict__ preds, const float* __restrict__ gts,
                            v2f* __restrict__ Pp, float* __restrict__ Np,
                            v2f* __restrict__ Pg, float* __restrict__ Ng) {
    int idx = blockIdx.x * blockDim.x + threadIdx.x;
    if (idx >= BATCH * NPTS) return;
    {
        float x = preds[3 * idx], y = preds[3 * idx + 1], z = preds[3 * idx + 2];
        float n = (x * x + y * y + z * z) * (1.0f / EPS_C);
        v2f lo; lo.x = x * SRT; lo.y = y * SRT;
        v2f hi; hi.x = z * SRT; hi.y = WLOG - n;
        Pp[2 * idx] = lo; Pp[2 * idx + 1] = hi; Np[idx] = n;
    }
    {
        float x = gts[3 * idx], y = gts[3 * idx + 1], z = gts[3 * idx + 2];
        float n = (x * x + y * y + z * z) * (1.0f / EPS_C);
        v2f lo; lo.x = x * SRT; lo.y = y * SRT;
        v2f hi; hi.x = z * SRT; hi.y = WLOG - n;
        Pg[2 * idx] = lo; Pg[2 * idx + 1] = hi; Ng[idx] = n;
    }
}

__global__ void init_run_kernel(float* __restrict__ Ad, float* __restrict__ Bd,
                                int* __restrict__ done) {
    int idx = blockIdx.x * blockDim.x + threadIdx.x;
    if (idx < BATCH * NPTS) { Ad[idx] = 0.0f; Bd[idx] = 0.0f; }
    if (idx == 0) *done = 0;
}

// ---------------------------------------------------------------
// Sinkhorn half-update:
// out[j] = n_y[j] - LSE_i( (2/eps)*x_i.y_j + dual[i] + wlog - n_x[i] )
// One WMMA f32 16x16x4 per 16x16 tile; dual folded into K3 via one branch-free fma.
// Two interleaved online-LSE chains for ILP; halves merged by shfl_xor(16).
// ---------------------------------------------------------------
__global__ __launch_bounds__(32) void lse_kernel(const v2f* __restrict__ PX,
                                                 const v2f* __restrict__ PY,
                                                 const float* __restrict__ NY,
                                                 const float* __restrict__ dualRed,
                                                 float* __restrict__ dualOut,
                                                 float* __restrict__ diff,
                                                 const int* __restrict__ done) {
    if (*done) return;
    const int b = blockIdx.y, jt = blockIdx.x, lane = threadIdx.x;
    const int nIdx = lane & 15;
    const bool hi = lane >= 16;
    const float hisel = hi ? 1.0f : 0.0f;
    const int j = jt * TILE + nIdx;

    v2f q = PY[2 * (b * NPTS + j) + (hi ? 1 : 0)];
    v2f bv; bv.x = q.x; bv.y = hi ? 1.0f : q.y;          // B K3-slot = 1

    const v2f*   AX = PX + 2 * (size_t)(b * NPTS) + (hi ? 1 : 0);
    const float* dR = dualRed + b * NPTS;

    float m0 = -INFINITY, s0 = 0.0f, m1 = -INFINITY, s1 = 0.0f;
    for (int it = 0; it < NT; it += 2) {
        {
            const int i = it * TILE + nIdx;
            v2f p = AX[2 * i];
            float du = dR[i];
            v2f a; a.x = p.x; a.y = fmaf(hisel, du, p.y);   // hi: (wlog-n)+du
            v8f d = wmma4(a, bv);
            float mn = fmaxf(m0, max8(d));
            float acc = s0 * __expf(m0 - mn);
#pragma unroll
            for (int k = 0; k < 8; ++k) acc += __expf(d[k] - mn);
            s0 = acc; m0 = mn;
        }
        {
            const int i = (it + 1) * TILE + nIdx;
            v2f p = AX[2 * i];
            float du = dR[i];
            v2f a; a.x = p.x; a.y = fmaf(hisel, du, p.y);
            v8f d = wmma4(a, bv);
            float mn = fmaxf(m1, max8(d));
            float acc = s1 * __expf(m1 - mn);
#pragma unroll
            for (int k = 0; k < 8; ++k) acc += __expf(d[k] - mn);
            s1 = acc; m1 = mn;
        }
    }
    // merge the two chains, then the two lane-halves of this column
    float M = fmaxf(m0, m1);
    float S = s0 * __expf(m0 - M) + s1 * __expf(m1 - M);
    const float M2 = __shfl_xor(M, 16, 32);
    const float S2 = __shfl_xor(S, 16, 32);
    const float Mf = fmaxf(M, M2);
    const float Sf = S * __expf(M - Mf) + S2 * __expf(M2 - Mf);
    const float lse = Mf + __logf(Sf);
    if (!hi) {
        const float outv = NY[b * NPTS + j] - lse;
        const float oldv = dualOut[b * NPTS + j];
        diff[b * NPTS + j]    = fabsf(oldv - outv);
        dualOut[b * NPTS + j] = outv;
    }
}

// deterministic error reduction + convergence flag (matches while_loop cond)
__global__ __launch_bounds__(256) void err_kernel(const float* __restrict__ diffA,
                                                  const float* __restrict__ diffB,
                                                  int* __restrict__ done) {
    if (*done) return;
    __shared__ float red[256];
    const int t = threadIdx.x;
    float aerr = 0.0f, berr = 0.0f;
    for (int b = 0; b < BATCH; ++b) {
        float s = 0.0f;
        for (int jj = t; jj < NPTS; jj += 256) s += diffA[b * NPTS + jj];
        red[t] = s; __syncthreads();
        for (int off = 128; off; off >>= 1) { if (t < off) red[t] += red[t + off]; __syncthreads(); }
        if (t == 0) aerr = fmaxf(aerr, EPS_C * red[0] / (float)NPTS);
        __syncthreads();
    }
    for (int b = 0; b < BATCH; ++b) {
        float s = 0.0f;
        for (int jj = t; jj < NPTS; jj += 256) s += diffB[b * NPTS + jj];
        red[t] = s; __syncthreads();
        for (int off = 128; off; off >>= 1) { if (t < off) red[t] += red[t + off]; __syncthreads(); }
        if (t == 0) berr = fmaxf(berr, EPS_C * red[0] / (float)NPTS);
        __syncthreads();
    }
    if (t == 0 && aerr < TOL_C && berr < TOL_C) *done = 1;
}

// ---------------------------------------------------------------
// cost = sum_ij exp(A[j]+B[i]-C/eps) * C, via two WMMAs sharing one B operand:
//   d1 = sgp + (B[i]-n_i)        (A1 = (s*x, B[i]-n_i))   arg = d1 + (A[j]-n_j)
//   d2 = -eps*sgp + eps*n_i      (A2 = (-eps*s*x, eps*n_i)) C  = d2 + eps*n_j
// ---------------------------------------------------------------
__global__ __launch_bounds__(32) void cost_kernel(const v2f* __restrict__ PX,
                                                  const v2f* __restrict__ PY,
                                                  const float* __restrict__ NY,
                                                  const float* __restrict__ dualRow,
                                                  const float* __restrict__ dualCol,
                                                  float* __restrict__ part) {
    const int b = blockIdx.y, jt = blockIdx.x, lane = threadIdx.x;
    const int nIdx = lane & 15;
    const bool hi = lane >= 16;
    const float hisel = hi ? 1.0f : 0.0f;
    const int j = jt * TILE + nIdx;

    v2f q = PY[2 * (b * NPTS + j) + (hi ? 1 : 0)];
    v2f bv; bv.x = q.x; bv.y = hi ? 1.0f : q.y;
    const float nj   = NY[b * NPTS + j];
    const float colu = dualCol[b * NPTS + j] - nj;   // A[j] - n_j
    const float coln = EPS_C * nj;                   // eps * n_j

    const v2f*   AX = PX + 2 * (size_t)(b * NPTS) + (hi ? 1 : 0);
    const float* dR = dualRow + b * NPTS;

    float acc0 = 0.0f, acc1 = 0.0f;
    for (int it = 0; it < NT; it += 2) {
#pragma unroll
        for (int u = 0; u < 2; ++u) {
            const int i = (it + u) * TILE + nIdx;
            v2f p = AX[2 * i];
            float du = dR[i];
            // A1: hi lane K3 = (wlog-n) + (du-wlog) = du - n
            v2f a1; a1.x = p.x;           a1.y = fmaf(hisel, du - WLOG, p.y);
            // A2: lo = -eps*(sx0,sx1); hi = (-eps*sx2, eps*n)
            v2f a2; a2.x = -EPS_C * p.x;  a2.y = fmaf(hisel, EPS_C * WLOG, -EPS_C * p.y);
            v8f d1 = wmma4(a1, bv);
            v8f d2 = wmma4(a2, bv);
            float a = 0.0f;
#pragma unroll
            for (int k = 0; k < 8; ++k) a += __expf(d1[k] + colu) * (d2[k] + coln);
            if (u == 0) acc0 += a; else acc1 += a;
        }
    }
    float acc = acc0 + acc1;
    for (int off = 16; off; off >>= 1) acc += __shfl_xor(acc, off, 32);
    if (lane == 0) part[b * NT + jt] = acc;
}

// chamfer: per output column j, min_i C(i,j) = eps*(n_j - max_i(sgp - n_i))
__global__ __launch_bounds__(32) void chamfer_kernel(const v2f* __restrict__ PX,
                                                     const v2f* __restrict__ PY,
                                                     const float* __restrict__ NY,
                                                     float* __restrict__ part) {
    const int b = blockIdx.y, jt = blockIdx.x, lane = threadIdx.x;
    const int nIdx = lane & 15;
    const bool hi = lane >= 16;
    const float hisel = hi ? 1.0f : 0.0f;
    const int j = jt * TILE + nIdx;

    v2f q = PY[2 * (b * NPTS + j) + (hi ? 1 : 0)];
    v2f bv; bv.x = q.x; bv.y = hi ? 1.0f : q.y;
    const v2f* AX = PX + 2 * (size_t)(b * NPTS) + (hi ? 1 : 0);

    float m0 = -INFINITY, m1 = -INFINITY;
    for (int it = 0; it < NT; it += 2) {
        {
            const int i = it * TILE + nIdx;
            v2f p = AX[2 * i];
            v2f a; a.x = p.x; a.y = fmaf(hisel, -WLOG, p.y);   // hi: -n_i
            m0 = fmaxf(m0, max8(wmma4(a, bv)));
        }
        {
            const int i = (it + 1) * TILE + nIdx;
            v2f p = AX[2 * i];
            v2f a; a.x = p.x; a.y = fmaf(hisel, -WLOG, p.y);
            m1 = fmaxf(m1, max8(wmma4(a, bv)));
        }
    }
    float m = fmaxf(m0, m1);
    m = fmaxf(m, __shfl_xor(m, 16, 32));
    float acc = EPS_C * (NY[b * NPTS + j] - m);   // min_i C(i,j); halves duplicated
    for (int off = 16; off; off >>= 1) acc += __shfl_xor(acc, off, 32);
    if (lane == 0) part[b * NT + jt] = acc * 0.5f;
}

__global__ void cost_reduce_kernel(const float* __restrict__ part, float* __restrict__ out) {
    const int b = threadIdx.x;
    if (b < BATCH) {
        float s = 0.0f;
        for (int t = 0; t < NT; ++t) s += part[b * NT + t];
        out[b] = s;
    }
}

__global__ void final_kernel(const float* __restrict__ cost1, const float* __restrict__ cost2,
                             const float* __restrict__ ch1, const float* __restrict__ ch2,
                             float* __restrict__ out) {
    const int t = threadIdx.x;
    const float inv_nm = 1.0f / ((float)NPTS * (float)NPTS);
    if (t < BATCH) out[t] = (cost1[t] - 0.5f * cost2[t]) * inv_nm;
    if (t == BATCH) {
        float ch = 0.0f;
        for (int b = 0; b < BATCH; ++b) {
            float s1 = 0.0f, s2 = 0.0f;
            for (int k = 0; k < NT; ++k) { s1 += ch1[b * NT + k]; s2 += ch2[b * NT + k]; }
            ch += (s1 + s2) / (float)NPTS;
        }
        out[BATCH] = ch * (1.0f / (float)BATCH);
    }
}

extern "C" void kernel_launch(void* const* d_in, const int* in_sizes, int n_in,
                              void* d_out, int out_size, void* d_ws, size_t ws_size,
                              hipStream_t stream) {
    const float* preds = (const float*)d_in[0];
    const float* gts   = (const float*)d_in[1];
    float* out = (float*)d_out;

    const size_t BN = (size_t)BATCH * NPTS;
    char* w = (char*)d_ws;
    size_t o = 0;
    v2f*   Pg   = (v2f*)(w + o);   o += BN * 2 * sizeof(v2f);   // 128 KB
    v2f*   Pp   = (v2f*)(w + o);   o += BN * 2 * sizeof(v2f);
    float* Ngr  = (float*)(w + o); o += BN * sizeof(float);
    float* Npr  = (float*)(w + o); o += BN * sizeof(float);
    float* Ad   = (float*)(w + o); o += BN * sizeof(float);
    float* Bd   = (float*)(w + o); o += BN * sizeof(float);
    float* diffA= (float*)(w + o); o += BN * sizeof(float);
    float* diffB= (float*)(w + o); o += BN * sizeof(float);
    float* part = (float*)(w + o); o += (size_t)BATCH * NT * sizeof(float);
    float* ch1  = (float*)(w + o); o += (size_t)BATCH * NT * sizeof(float);
    float* ch2  = (float*)(w + o); o += (size_t)BATCH * NT * sizeof(float);
    float* cost1= (float*)(w + o); o += 16;
    float* cost2= (float*)(w + o); o += 16;
    int*   done = (int*)(w + o);   o += 16;

    const dim3 gridT(NT, BATCH);
    const dim3 blk32(32);

    prep_kernel<<<(BATCH * NPTS + 255) / 256, 256, 0, stream>>>(preds, gts, Pp, Npr, Pg, Ngr);

    // chamfer: loss_1 (min over gt rows, per pred col) + loss_2 (min over preds, per gt)
    chamfer_kernel<<<gridT, blk32, 0, stream>>>(Pg, Pp, Npr, ch1);
    chamfer_kernel<<<gridT, blk32, 0, stream>>>(Pp, Pg, Ngr, ch2);

    // ---- Sinkhorn run 1: C(gt, pred) ----
    init_run_kernel<<<(BATCH * NPTS + 255) / 256, 256, 0, stream>>>(Ad, Bd, done);
    for (int it = 0; it < NITS; ++it) {
        lse_kernel<<<gridT, blk32, 0, stream>>>(Pg, Pp, Npr, Bd, Ad, diffA, done); // A update
        lse_kernel<<<gridT, blk32, 0, stream>>>(Pp, Pg, Ngr, Ad, Bd, diffB, done); // B update
        err_kernel<<<1, 256, 0, stream>>>(diffA, diffB, done);
    }
    cost_kernel<<<gridT, blk32, 0, stream>>>(Pg, Pp, Npr, Bd, Ad, part);
    cost_reduce_kernel<<<1, 32, 0, stream>>>(part, cost1);

    // ---- Sinkhorn run 2: C(pred, pred) entropy correction ----
    init_run_kernel<<<(BATCH * NPTS + 255) / 256, 256, 0, stream>>>(Ad, Bd, done);
    for (int it = 0; it < NITS; ++it) {
        lse_kernel<<<gridT, blk32, 0, stream>>>(Pp, Pp, Npr, Bd, Ad, diffA, done);
        lse_kernel<<<gridT, blk32, 0, stream>>>(Pp, Pp, Npr, Ad, Bd, diffB, done);
        err_kernel<<<1, 256, 0, stream>>>(diffA, diffB, done);
    }
    cost_kernel<<<gridT, blk32, 0, stream>>>(Pp, Pp, Npr, Bd, Ad, part);
    cost_reduce_kernel<<<1, 32, 0, stream>>>(part, cost2);

    final_kernel<<<1, 32, 0, stream>>>(cost1, cost2, ch1, ch2, out);
}